// Discriminiator_84593675862379
// MI455X (gfx1250) — compile-verified
//
#include <hip/hip_runtime.h>
#include <hip/hip_bf16.h>

typedef float v2f __attribute__((ext_vector_type(2)));
typedef float v8f __attribute__((ext_vector_type(8)));

#define BB 256
#define SS 2048
#define II 128
#define NIN 40   // LSTM_IN
#define NG 80    // 4*H
#define HH 20

// ---------------------------------------------------------------------------
// Kernel 1: GX[s,b,g] = relu(feed[b,s,:] @ W1^T + b1) @ W_ih^T + (b_ih + b_hh)
// Uses V_WMMA_F32_16X16X4_F32 (exact f32 matrix op) for both GEMMs.
// One wave handles a 16-row tile (16 consecutive b for a fixed s).
// ---------------------------------------------------------------------------
__global__ __launch_bounds__(128) void disc_gemm_kernel(
    const float* __restrict__ feed, const float* __restrict__ W1,
    const float* __restrict__ b1,   const float* __restrict__ Wih,
    const float* __restrict__ bih,  const float* __restrict__ bhh,
    float* __restrict__ gx)
{
    // B-fragments packed in WMMA lane order:
    //   idx = ((kk*NT + nt)*32 + lane)*2 + j ;  K = kk*4 + 2*(lane>>4) + j ; N = nt*16 + (lane&15)
    __shared__ float sW1B[32 * 3 * 32 * 2];   // 6144 f32 : K=128, N padded 40->48
    __shared__ float sWihB[12 * 5 * 32 * 2];  // 3840 f32 : K padded 40->48, N=80
    __shared__ float sb1[48];
    __shared__ float sbsum[80];
    __shared__ float sX1[4][832];             // per-wave X1 tile, [K=48][M=16] pad 17

    const int tid = threadIdx.x;

    for (int idx = tid; idx < 6144; idx += 128) {
        int j = idx & 1, lane = (idx >> 1) & 31, rest = idx >> 6;
        int nt = rest % 3, kk = rest / 3;
        int n = nt * 16 + (lane & 15);
        int k = kk * 4 + ((lane >> 4) << 1) + j;
        sW1B[idx] = (n < NIN) ? W1[n * II + k] : 0.0f;
    }
    for (int idx = tid; idx < 3840; idx += 128) {
        int j = idx & 1, lane = (idx >> 1) & 31, rest = idx >> 6;
        int nt = rest % 5, kk = rest / 5;
        int n = nt * 16 + (lane & 15);
        int k = kk * 4 + ((lane >> 4) << 1) + j;
        sWihB[idx] = (k < NIN) ? Wih[n * NIN + k] : 0.0f;
    }
    for (int idx = tid; idx < 48; idx += 128) sb1[idx]  = (idx < NIN) ? b1[idx] : 0.0f;
    for (int idx = tid; idx < 80; idx += 128) sbsum[idx] = bih[idx] + bhh[idx];
    __syncthreads();

    const int wid  = tid >> 5;
    const int lane = tid & 31;
    const int l15  = lane & 15;
    const int hi   = lane >> 4;           // half-wave select (K pair {0,1} vs {2,3})
    const int NTILES = (SS * BB) / 16;    // 32768
    const int gwave  = blockIdx.x * 4 + wid;
    const int nwaves = gridDim.x * 4;

    for (int tile = gwave; tile < NTILES; tile += nwaves) {
        const int s  = tile >> 4;
        const int b0 = (tile & 15) << 4;
        // A-fragment row for this lane: M = lane&15
        const float* arow = feed + ((size_t)(b0 + l15) * SS + s) * II;

        // Prefetch this wave's next tile row (global_prefetch_b8).
        int tn = tile + nwaves;
        if (tn < NTILES) {
            const float* prow = feed + ((size_t)(((tn & 15) << 4) + l15) * SS + (tn >> 4)) * II;
            __builtin_prefetch(prow, 0, 1);
        }

        // ---- GEMM 1: X1[16x48] = A[16x128] * W1^T ----
        v8f acc[3] = {};
        const v2f* w1v = (const v2f*)sW1B;
        #pragma unroll 4
        for (int kk = 0; kk < 32; ++kk) {
            v2f a = *(const v2f*)(arow + kk * 4 + (hi << 1));
            const v2f* bp = w1v + (kk * 3) * 32 + lane;
            #pragma unroll
            for (int nt = 0; nt < 3; ++nt) {
                acc[nt] = __builtin_amdgcn_wmma_f32_16x16x4_f32(
                    false, a, false, bp[nt * 32], (short)0, acc[nt], false, false);
            }
        }

        // bias + relu; transpose C-layout -> A-fragment layout via LDS [K][M] (pad 17)
        #pragma unroll
        for (int nt = 0; nt < 3; ++nt) {
            float bv = sb1[nt * 16 + l15];      // column N of this lane
            #pragma unroll
            for (int r = 0; r < 8; ++r) {
                int M = r + (hi << 3);
                float x = fmaxf(acc[nt][r] + bv, 0.0f);
                sX1[wid][(nt * 16 + l15) * 17 + M] = x;   // N becomes K of GEMM 2
            }
        }

        // ---- GEMM 2: GX[16x80] = X1[16x48] * Wih^T ----
        v8f d[5] = {};
        const v2f* wiv = (const v2f*)sWihB;
        #pragma unroll 3
        for (int kk = 0; kk < 12; ++kk) {
            int K = kk * 4 + (hi << 1);
            v2f a2;
            a2.x = sX1[wid][K * 17 + l15];
            a2.y = sX1[wid][(K + 1) * 17 + l15];
            const v2f* bp = wiv + (kk * 5) * 32 + lane;
            #pragma unroll
            for (int nt = 0; nt < 5; ++nt) {
                d[nt] = __builtin_amdgcn_wmma_f32_16x16x4_f32(
                    false, a2, false, bp[nt * 32], (short)0, d[nt], false, false);
            }
        }

        // store GX (time-major [S,B,80]); C-layout: M = r + 8*hi, N = l15
        const int mglob = tile * 16;
        #pragma unroll
        for (int nt = 0; nt < 5; ++nt) {
            float bs = sbsum[nt * 16 + l15];
            #pragma unroll
            for (int r = 0; r < 8; ++r) {
                int M = r + (hi << 3);
                gx[(size_t)(mglob + M) * NG + nt * 16 + l15] = d[nt][r] + bs;
            }
        }
    }
}

// ---------------------------------------------------------------------------
// Kernel 2: sequential LSTM scan + time-mean + final 20->2 projection.
// 320 threads = 4 batch rows x 80 gate-threads; h/c state in LDS/registers.
// ---------------------------------------------------------------------------
__global__ __launch_bounds__(320) void disc_lstm_kernel(
    const float* __restrict__ gx,  const float* __restrict__ Whh,
    const float* __restrict__ Wf,  const float* __restrict__ bfv,
    const float* __restrict__ h0,  const float* __restrict__ c0,
    float* __restrict__ out)
{
    const int tid = threadIdx.x;
    const int bl  = tid / 80;            // 0..3 local batch row
    const int g   = tid - bl * 80;       // 0..79 gate index (i,f,g,o blocks of 20)
    const int b   = blockIdx.x * 4 + bl;

    __shared__ float hsh[4][HH];
    __shared__ float gact[4][NG];
    __shared__ float hsum_sh[4][HH];

    float whh[HH];
    #pragma unroll
    for (int k = 0; k < HH; ++k) whh[k] = Whh[g * HH + k];

    float c = 0.0f, hsum = 0.0f;
    if (g < HH) { hsh[bl][g] = h0[b * HH + g]; c = c0[b * HH + g]; }
    __syncthreads();

    const float* gxp = gx + (size_t)b * NG + g;
    float cur = gxp[0];

    for (int t = 0; t < SS; ++t) {
        float nxt = 0.0f;
        if (t + 1 < SS) nxt = gxp[(size_t)(t + 1) * (BB * NG)];   // double-buffered stream

        float pre = cur;
        #pragma unroll
        for (int k = 0; k < HH; ++k) pre += hsh[bl][k] * whh[k];

        float a;
        if (g >= 40 && g < 60) a = tanhf(pre);                    // cell gate
        else                   a = 1.0f / (1.0f + __expf(-pre));  // i, f, o
        gact[bl][g] = a;
        __syncthreads();

        if (g < HH) {
            float ig = gact[bl][g], fg = gact[bl][g + 20];
            float gg = gact[bl][g + 40], og = gact[bl][g + 60];
            c = fg * c + ig * gg;
            float h = og * tanhf(c);
            hsh[bl][g] = h;
            hsum += h;
        }
        __syncthreads();
        cur = nxt;
    }

    if (g < HH) hsum_sh[bl][g] = hsum;
    __syncthreads();

    if (g < 2) {
        float acc = bfv[g];
        const float inv = 1.0f / (float)SS;
        #pragma unroll
        for (int k = 0; k < HH; ++k) acc += (hsum_sh[bl][k] * inv) * Wf[g * HH + k];
        out[b * 2 + g] = acc;
    }
}

// ---------------------------------------------------------------------------
extern "C" void kernel_launch(void* const* d_in, const int* in_sizes, int n_in,
                              void* d_out, int out_size, void* d_ws, size_t ws_size,
                              hipStream_t stream) {
    const float* feed = (const float*)d_in[0];
    const float* W1   = (const float*)d_in[1];
    const float* b1   = (const float*)d_in[2];
    const float* Wih  = (const float*)d_in[3];
    const float* Whh  = (const float*)d_in[4];
    const float* bih  = (const float*)d_in[5];
    const float* bhh  = (const float*)d_in[6];
    const float* Wf   = (const float*)d_in[7];
    const float* bfv  = (const float*)d_in[8];
    const float* h0   = (const float*)d_in[9];
    const float* c0   = (const float*)d_in[10];
    float* out = (float*)d_out;

    float* gx = (float*)d_ws;   // [S, B, 80] f32 = 160 MB scratch

    disc_gemm_kernel<<<2048, 128, 0, stream>>>(feed, W1, b1, Wih, bih, bhh, gx);
    disc_lstm_kernel<<<BB / 4, 320, 0, stream>>>(gx, Whh, Wf, bfv, h0, c0, out);
}